// DeformConv2d_71631464562706
// MI455X (gfx1250) — compile-verified
//
#include <hip/hip_runtime.h>
#include <hip/hip_bf16.h>

typedef __attribute__((ext_vector_type(2))) float v2f;
typedef __attribute__((ext_vector_type(8))) float v8f;

#define BATCH 2
#define CIN   64
#define HIN   128
#define WIN   128
#define NPT_  9
#define OUTC  128
#define HP    130      // padded
#define HE    384      // expanded (3*H)
#define OH    382      // final conv out (384-3+1), stride 1 pad 0
#define OW    382
#define KTOT  576      // 64*9
#define KPAD  68       // LDS row stride (floats) to split bank groups

// ---------------- Stage 1: offset (18ch) + modulation (9ch) conv ----------------
__global__ void dcn_offset_mod(const float* __restrict__ x,
                               const float* __restrict__ p_w, const float* __restrict__ p_b,
                               const float* __restrict__ m_w, const float* __restrict__ m_b,
                               float* __restrict__ offs, float* __restrict__ mod)
{
    int idx = blockIdx.x * blockDim.x + threadIdx.x;
    int total = BATCH * 27 * HIN * WIN;
    if (idx >= total) return;
    int j = idx % WIN; int t = idx / WIN;
    int i = t % HIN;   t /= HIN;
    int ch = t % 27;   int b = t / 27;

    const float* wgt; float bias;
    if (ch < 18) { wgt = p_w + ch * CIN * 9;        bias = p_b[ch]; }
    else         { wgt = m_w + (ch - 18) * CIN * 9; bias = m_b[ch - 18]; }

    float acc = bias;
    for (int ic = 0; ic < CIN; ++ic) {
        const float* xc = x + (size_t)(b * CIN + ic) * HIN * WIN;
        const float* wc = wgt + ic * 9;
        #pragma unroll
        for (int kh = 0; kh < 3; ++kh) {
            int ii = i + kh - 1;
            if (ii < 0 || ii >= HIN) continue;
            #pragma unroll
            for (int kw = 0; kw < 3; ++kw) {
                int jj = j + kw - 1;
                if (jj < 0 || jj >= WIN) continue;
                acc = fmaf(xc[ii * WIN + jj], wc[kh * 3 + kw], acc);
            }
        }
    }
    if (ch < 18) offs[((b * 18 + ch) * HIN + i) * WIN + j] = acc;
    else         mod [((b * 9 + (ch - 18)) * HIN + i) * WIN + j] = 1.0f / (1.0f + __expf(-acc));
}

// ---------------- Stage 2: reorder conv_w -> W2T[n][k], k=(kh*3+kw)*64+ic ----------------
__global__ void dcn_reorder_w(const float* __restrict__ conv_w, float* __restrict__ w2t)
{
    int idx = blockIdx.x * blockDim.x + threadIdx.x;
    if (idx >= OUTC * KTOT) return;
    int k = idx % KTOT; int n = idx / KTOT;
    int khkw = k >> 6;  int ic = k & 63;
    w2t[n * KTOT + k] = conv_w[((n * CIN + ic) * 3 + khkw / 3) * 3 + (khkw % 3)];
}

// ---------------- Stage 3: modulated bilinear sampling -> x_off NHWC (B,384,384,64) ----------------
__global__ void dcn_sample(const float* __restrict__ x,
                           const float* __restrict__ offs,
                           const float* __restrict__ mod,
                           float* __restrict__ xoff)
{
    int blk = blockIdx.x;                 // (b, i, j)
    int j = blk % WIN; int t = blk / WIN;
    int i = t % HIN;   int b = t / HIN;

    for (int e = threadIdx.x; e < NPT_ * CIN; e += blockDim.x) {
        int n = e >> 6; int c = e & 63;
        int a = n / 3, bb = n % 3;

        float offx = offs[((b * 18 + n) * HIN + i) * WIN + j];
        float offy = offs[((b * 18 + 9 + n) * HIN + i) * WIN + j];
        float px = (float)(i + a) + offx;   // (i+1) + (a-1) + off
        float py = (float)(j + bb) + offy;

        float fx = floorf(px), fy = floorf(py);
        float pxc = fminf(fmaxf(px, 0.f), (float)(HP - 1));
        float pyc = fminf(fmaxf(py, 0.f), (float)(HP - 1));
        int qlx = (int)fminf(fmaxf(fx,       0.f), (float)(HP - 1));
        int qly = (int)fminf(fmaxf(fy,       0.f), (float)(HP - 1));
        int qrx = (int)fminf(fmaxf(fx + 1.f, 0.f), (float)(HP - 1));
        int qry = (int)fminf(fmaxf(fy + 1.f, 0.f), (float)(HP - 1));

        float glt = (1.f + ((float)qlx - pxc)) * (1.f + ((float)qly - pyc));
        float grb = (1.f - ((float)qrx - pxc)) * (1.f - ((float)qry - pyc));
        float glb = (1.f + ((float)qlx - pxc)) * (1.f - ((float)qry - pyc));
        float grt = (1.f - ((float)qrx - pxc)) * (1.f + ((float)qly - pyc));

        const float* xc = x + (size_t)(b * CIN + c) * HIN * WIN;
        auto fetch = [&](int qx, int qy) -> float {
            int ii = qx - 1, jj = qy - 1;   // padded -> unpadded coords, zeros outside
            if (ii < 0 || ii >= HIN || jj < 0 || jj >= WIN) return 0.f;
            return xc[ii * WIN + jj];
        };
        float v = glt * fetch(qlx, qly) + grb * fetch(qrx, qry)
                + glb * fetch(qlx, qry) + grt * fetch(qrx, qly);
        v *= mod[((b * 9 + n) * HIN + i) * WIN + j];

        xoff[((size_t)(b * HE + (3 * i + a)) * HE + (3 * j + bb)) * CIN + c] = v;
    }
}

// ---------------- Stage 4: implicit-GEMM 3x3 conv via V_WMMA_F32_16X16X4_F32 ----------------
// Wave tile: M=16 (ow), N=128 (all oc), fixed (b, oh). K = 9 chunks of 64 channels,
// each chunk == one (kh,kw) tap so A addressing is a single spatial offset.
__global__ __launch_bounds__(256) void dcn_gemm(const float* __restrict__ xoff,
                                                const float* __restrict__ w2t,
                                                float* __restrict__ out)
{
    __shared__ float blds[OUTC * KPAD];   // 128*68*4 = 34,816 B

    const int oh   = blockIdx.y;
    const int b    = blockIdx.z;
    const int wave = threadIdx.x >> 5;
    const int lane = threadIdx.x & 31;
    const int l15  = lane & 15;
    const int hi   = lane >> 4;           // 0: K0/K1 half, 1: K2/K3 half
    const int ow0  = blockIdx.x * 128 + wave * 16;

    // clamped M position for loads (stores are masked later; EXEC stays all-1s for WMMA)
    int owm = ow0 + l15;
    int owl = owm < (OW - 1) ? owm : (OW - 1);

    v8f acc[8];
    #pragma unroll
    for (int nt = 0; nt < 8; ++nt)
        acc[nt] = v8f{0.f, 0.f, 0.f, 0.f, 0.f, 0.f, 0.f, 0.f};

    for (int kc = 0; kc < 9; ++kc) {      // (kh,kw) tap == K chunk of 64
        const int kh = kc / 3, kw = kc % 3;
        __syncthreads();
        for (int e = threadIdx.x; e < OUTC * 64; e += 256) {
            int n = e >> 6, kk = e & 63;
            blds[n * KPAD + kk] = w2t[n * KTOT + kc * 64 + kk];
        }
        __syncthreads();

        const float* aptr = xoff + ((size_t)(b * HE + (oh + kh)) * HE + (owl + kw)) * CIN + hi * 2;
        for (int ks = 0; ks < 64; ks += 4) {
            v2f afrag = *(const v2f*)(aptr + ks);      // A 16x4: lane<16 -> K0,K1 ; lane>=16 -> K2,K3
            #pragma unroll
            for (int nt = 0; nt < 8; ++nt) {
                v2f bfrag = *(const v2f*)(&blds[(nt * 16 + l15) * KPAD + ks + hi * 2]);
                acc[nt] = __builtin_amdgcn_wmma_f32_16x16x4_f32(
                    /*neg_a=*/false, afrag, /*neg_b=*/false, bfrag,
                    /*c_mod=*/(short)0, acc[nt], /*reuse_a=*/false, /*reuse_b=*/false);
            }
        }
    }

    // D layout (16x16 f32): VGPR r -> M=r (lanes 0-15) / M=r+8 (lanes 16-31), N=lane&15
    #pragma unroll
    for (int nt = 0; nt < 8; ++nt) {
        int oc = nt * 16 + l15;
        #pragma unroll
        for (int r = 0; r < 8; ++r) {
            int ow = ow0 + r + hi * 8;
            if (ow < OW)
                out[(((size_t)b * OUTC + oc) * OH + oh) * OW + ow] = acc[nt][r];
        }
    }
}

extern "C" void kernel_launch(void* const* d_in, const int* in_sizes, int n_in,
                              void* d_out, int out_size, void* d_ws, size_t ws_size,
                              hipStream_t stream) {
    const float* x      = (const float*)d_in[0];
    const float* p_w    = (const float*)d_in[1];
    const float* p_b    = (const float*)d_in[2];
    const float* m_w    = (const float*)d_in[3];
    const float* m_b    = (const float*)d_in[4];
    const float* conv_w = (const float*)d_in[5];
    float* out = (float*)d_out;

    float* ws   = (float*)d_ws;
    float* offs = ws;                                   // 2*18*128*128 = 589,824 f
    float* mod  = offs + 2 * 18 * HIN * WIN;            // 2*9*128*128  = 294,912 f
    float* w2t  = mod  + 2 * 9 * HIN * WIN;             // 128*576      =  73,728 f
    float* xoff = w2t  + OUTC * KTOT;                   // 2*384*384*64 = 18,874,368 f

    {
        int total = BATCH * 27 * HIN * WIN;
        dcn_offset_mod<<<(total + 255) / 256, 256, 0, stream>>>(x, p_w, p_b, m_w, m_b, offs, mod);
    }
    {
        int total = OUTC * KTOT;
        dcn_reorder_w<<<(total + 255) / 256, 256, 0, stream>>>(conv_w, w2t);
    }
    {
        dcn_sample<<<BATCH * HIN * WIN, 256, 0, stream>>>(x, offs, mod, xoff);
    }
    {
        dim3 grid(3, OH, BATCH);   // 3*128 = 384 >= 382 ow columns per row
        dcn_gemm<<<grid, 256, 0, stream>>>(xoff, w2t, out);
    }
}